// ComputeIdsLayer_58188216926857
// MI455X (gfx1250) — compile-verified
//
#include <hip/hip_runtime.h>

// Problem constants (fixed by the reference harness).
#define NID   128        // one-hot width
#define SPOS  2048       // sequence length S
#define BLK   1024       // threads per block (32 wave32 waves)
#define NWAVES (BLK / 32)

// ---- gfx1250 feature detection (compile-only loop: guard everything) ----
#if defined(__has_builtin)
# if __has_builtin(__builtin_amdgcn_global_load_async_to_lds_b128) && \
     __has_builtin(__builtin_amdgcn_s_wait_asynccnt)
#  define USE_ASYNC_LDS 1
# endif
#endif
#ifndef USE_ASYNC_LDS
# define USE_ASYNC_LDS 0
#endif

// b128 async op moves one int 4-vector; builtin takes AS1/AS3 v4i pointers
// (param type confirmed by the round-1 clang diagnostic).
typedef int   v4i __attribute__((vector_size(16)));
typedef float v4f __attribute__((ext_vector_type(4)));   // native vector: OK for nontemporal builtin
typedef __attribute__((address_space(1))) void  GVv;   // global void
typedef __attribute__((address_space(3))) void  LVv;   // LDS void
typedef __attribute__((address_space(1))) v4i   GV4;   // global v4i
typedef __attribute__((address_space(3))) v4i   LV4;   // LDS v4i

__device__ __forceinline__ unsigned ballot32(bool p) {
#if defined(__has_builtin) && __has_builtin(__builtin_amdgcn_ballot_w32)
    return __builtin_amdgcn_ballot_w32(p);
#else
    return (unsigned)__ballot(p);
#endif
}

// One block per batch row.
__global__ __launch_bounds__(BLK)
void new_id_onehot_kernel(const int*   __restrict__ enref_ids,   // [B, S]
                          const int*   __restrict__ seq_len,     // [B]
                          const float* __restrict__ logits,      // [B, S, 2]
                          float*       __restrict__ out)         // [B, S, NID]
{
    const int b    = blockIdx.x;
    const int tid  = threadIdx.x;
    const int lane = tid & 31;
    const int wave = tid >> 5;

    __shared__ unsigned      usedMask[NID / 32];   // 128-bit used-id mask
    __shared__ unsigned char newid[NID];           // k -> assigned id (k-th free id, 0 after exhaustion)
    __shared__ short         rowid[SPOS];          // per-row one-hot column, -1 = all-zero row
    __shared__ int           waveSum[NWAVES];
    __shared__ int           waveBase[NWAVES];
#if USE_ASYNC_LDS
    __shared__ int           idsLds[SPOS];         // async-DMA staged enref_ids row
#endif

    if (tid < NID / 32) usedMask[tid] = 0u;

    const int  len    = seq_len[b];
    const int* idsRow = enref_ids + (size_t)b * SPOS;

#if USE_ASYNC_LDS
    // DMA the whole 8KB id row into LDS (b128 = 4 ints per op, 512 ops).
    if (tid < SPOS / 4) {
        __builtin_amdgcn_global_load_async_to_lds_b128(
            (GV4*)(GVv*)(idsRow + tid * 4),
            (LV4*)(LVv*)(&idsLds[tid * 4]),
            /*offset=*/0, /*cpol=*/0);
    }
#endif

    // Overlap with the DMA: flags for s0 = 2*tid, s1 = 2*tid+1.
    // logits row viewed as float4: .x = logit[s0][0], .z = logit[s1][0].
    const float4* lg = (const float4*)(logits + (size_t)b * SPOS * 2);
    const float4  v  = lg[tid];
    const bool f0 = v.x > 0.0f;
    const bool f1 = v.z > 0.0f;

    // wave32-native exclusive prefix of flag counts.
    const unsigned bal0 = ballot32(f0);
    const unsigned bal1 = ballot32(f1);
    const int pre = (int)__builtin_amdgcn_mbcnt_lo(bal0, 0u)
                  + (int)__builtin_amdgcn_mbcnt_lo(bal1, 0u);
    if (lane == 0) waveSum[wave] = __popc(bal0) + __popc(bal1);

#if USE_ASYNC_LDS
    __builtin_amdgcn_s_wait_asynccnt(0);
#endif
    __syncthreads();   // usedMask init, waveSum, staged ids all visible

    // Phase 1: mark used ids with LDS atomics (ds_or_b32).
    for (int s = tid; s < len; s += BLK) {
#if USE_ASYNC_LDS
        int id = idsLds[s];
#else
        int id = idsRow[s];
#endif
        id &= (NID - 1);
        atomicOr(&usedMask[id >> 5], 1u << (id & 31));
    }
    // Concurrent: scan the 32 wave totals (trivial serial work).
    if (tid == 0) {
        int acc = 0;
        for (int w = 0; w < NWAVES; ++w) { waveBase[w] = acc; acc += waveSum[w]; }
    }
    __syncthreads();   // usedMask complete

    // Phase 2: k-th free id table; argmax(~all_used) == 0 => fill tail with 0.
    if (tid == 0) {
        int cnt = 0;
        for (int n = 0; n < NID; ++n)
            if (!((usedMask[n >> 5] >> (n & 31)) & 1u))
                newid[cnt++] = (unsigned char)n;
        for (; cnt < NID; ++cnt) newid[cnt] = 0;
    }
    __syncthreads();   // newid + waveBase ready

    // Phase 3: resolve each row's one-hot column.
    {
        const int k0 = waveBase[wave] + pre;
        const int k1 = k0 + (f0 ? 1 : 0);
        const int s0 = 2 * tid;
        rowid[s0]     = f0 ? (short)(k0 < NID ? newid[k0] : 0) : (short)-1;
        rowid[s0 + 1] = f1 ? (short)(k1 < NID ? newid[k1] : 0) : (short)-1;
    }
    __syncthreads();

    // Phase 4: stream 1 MB of output. Wave w writes rows w, w+32, ...
    // Each b128 store covers a full contiguous 512B row per wave-instruction.
    // Output (268 MB/call) is never re-read and exceeds the 192 MB L2 ->
    // use non-temporal stores so the dead write stream doesn't thrash L2.
    float* outBase = out + (size_t)b * SPOS * NID;
    for (int s = wave; s < SPOS; s += NWAVES) {
        const int rid = rowid[s];                  // wave-uniform LDS broadcast
        v4f val = {0.0f, 0.0f, 0.0f, 0.0f};
        if ((rid >> 2) == lane) {                  // rid in [4*lane, 4*lane+4)
            ((float*)&val)[rid & 3] = 1.0f;
        }
        __builtin_nontemporal_store(val, (v4f*)(outBase + (size_t)s * NID) + lane);
    }
}

extern "C" void kernel_launch(void* const* d_in, const int* in_sizes, int n_in,
                              void* d_out, int out_size, void* d_ws, size_t ws_size,
                              hipStream_t stream) {
    const int*   enref_ids = (const int*)d_in[0];   // [B, S] int32
    const int*   seq_len   = (const int*)d_in[1];   // [B]    int32
    const float* logits    = (const float*)d_in[2]; // [B, S, 2] f32
    float*       out       = (float*)d_out;         // [B, S, 128] f32

    const int B = in_sizes[1];                      // 256
    new_id_onehot_kernel<<<B, BLK, 0, stream>>>(enref_ids, seq_len, logits, out);
}